// GuidedFilter2d_23562190586120
// MI455X (gfx1250) — compile-verified
//
#include <hip/hip_runtime.h>

// Guided filter (color guide), B=4, C=4, G=3, H=W=1024, r=8, eps=1e-4.
// Two-kernel fused design (roofline ~0.74 GB HBM traffic @ 23.3 TB/s).
// 17-tap 2D box filter of each 16x16 tile is computed as
//   Out = Bv(16x32) * X(32x32) * Bh(32x16)
// with 0/1 banded matrices via V_WMMA_F32_16X16X4_F32 (CDNA5 wave32 WMMA).

typedef float v2f __attribute__((ext_vector_type(2)));
typedef float v8f __attribute__((ext_vector_type(8)));

#define RAD   8
#define EPSV  1e-4f
#define HD    1024
#define WD    1024
#define HW    (1024 * 1024)
#define TILE  16
#define HALO  32          // TILE + 2*RAD
#define HSTR  33          // padded LDS stride for 32-wide buffers

__device__ __forceinline__ float nlen(int i) {
    int lo = i - RAD; if (lo < 0) lo = 0;
    int hi = i + RAD; if (hi > HD - 1) hi = HD - 1;
    return (float)(hi - lo + 1);
}

// 2D box filter of one 16x16 tile from a 32x32 zero-padded halo source.
// One full wave32 executes this; `lt` is a wave-private 16x33 LDS scratch.
// Returns the boxed tile in WMMA C/D layout:
//   lane holds rows (v + (lane<16?0:8)), col (lane&15), v = 0..7.
template <typename F>
__device__ __forceinline__ v8f box2d_wmma(int lane, float* lt, F srcAt) {
    const int half = lane >> 4;      // 0: lanes 0-15, 1: lanes 16-31
    const int l15  = lane & 15;

    // ---- vertical pass: T(16x32) = Bv(16x32) * X(32x32) ----
    // Bv[m][k] = 1 iff m <= k <= m+16  (out row m, halo row k)
#pragma unroll
    for (int j = 0; j < 2; ++j) {                 // 16-col blocks of T
        v8f acc = {0.f, 0.f, 0.f, 0.f, 0.f, 0.f, 0.f, 0.f};
#pragma unroll
        for (int kb = 0; kb < 2; ++kb) {          // 16-row K blocks
#pragma unroll
            for (int k4 = 0; k4 < 4; ++k4) {      // K=4 sub-steps
                const int k0 = kb * 16 + k4 * 4 + half * 2;
                const int m  = l15;
                v2f A, Bx;
                A.x = (k0     >= m && k0     <= m + 16) ? 1.f : 0.f;
                A.y = (k0 + 1 >= m && k0 + 1 <= m + 16) ? 1.f : 0.f;
                const int n = j * 16 + l15;
                Bx.x = srcAt(k0,     n);
                Bx.y = srcAt(k0 + 1, n);
                acc = __builtin_amdgcn_wmma_f32_16x16x4_f32(
                          false, A, false, Bx, (short)0, acc, false, false);
            }
        }
#pragma unroll
        for (int v = 0; v < 8; ++v)
            lt[(v + half * 8) * HSTR + j * 16 + l15] = acc[v];
    }

    // ---- horizontal pass: Out(16x16) = T(16x32) * Bh(32x16) ----
    // Bh[k][n] = 1 iff n <= k <= n+16  (halo col k, out col n)
    v8f out = {0.f, 0.f, 0.f, 0.f, 0.f, 0.f, 0.f, 0.f};
#pragma unroll
    for (int kb = 0; kb < 2; ++kb) {
#pragma unroll
        for (int k4 = 0; k4 < 4; ++k4) {
            const int k0 = kb * 16 + k4 * 4 + half * 2;
            const int m  = l15;
            v2f A, Bb;
            A.x = lt[m * HSTR + k0];
            A.y = lt[m * HSTR + k0 + 1];
            const int n = l15;
            Bb.x = (k0     >= n && k0     <= n + 16) ? 1.f : 0.f;
            Bb.y = (k0 + 1 >= n && k0 + 1 <= n + 16) ? 1.f : 0.f;
            out = __builtin_amdgcn_wmma_f32_16x16x4_f32(
                      false, A, false, Bb, (short)0, out, false, false);
        }
    }
    return out;
}

// ---------------------------------------------------------------------------
// K1: box-filter the 25 statistics planes for one 16x16 tile, then do the
//     per-pixel 3x3 covariance solve and emit a (12 planes) + b (4 planes).
// ---------------------------------------------------------------------------
__global__ __launch_bounds__(128) void gf_stats_solve(
        const float* __restrict__ x, const float* __restrict__ g,
        float* __restrict__ ab) {
    __shared__ float Gs[3][HALO * HSTR];   // guide halo tiles
    __shared__ float Ss[4][HALO * HSTR];   // src halo tiles
    __shared__ float Ms[25][TILE * TILE];  // boxed (undivided) statistics
    __shared__ float Ts[4][TILE * HSTR];   // per-wave WMMA scratch

    const int tid  = threadIdx.x;
    const int lane = tid & 31;
    const int wave = tid >> 5;
    const int half = lane >> 4;
    const int l15  = lane & 15;
    const int tx0  = blockIdx.x * TILE;
    const int ty0  = blockIdx.y * TILE;
    const int bz   = blockIdx.z;

    const float* gp = g + (size_t)bz * 3 * HW;
    const float* xp = x + (size_t)bz * 4 * HW;

    // cooperative zero-padded halo load (7 planes x 32x32)
    for (int idx = tid; idx < 7 * HALO * HALO; idx += 128) {
        const int p  = idx >> 10;
        const int r  = (idx >> 5) & 31;
        const int cc = idx & 31;
        const int gy = ty0 - RAD + r;
        const int gx = tx0 - RAD + cc;
        float v = 0.f;
        if (gy >= 0 && gy < HD && gx >= 0 && gx < WD) {
            v = (p < 3) ? gp[(size_t)p * HW + gy * WD + gx]
                        : xp[(size_t)(p - 3) * HW + gy * WD + gx];
        }
        if (p < 3) Gs[p][r * HSTR + cc] = v;
        else       Ss[p - 3][r * HSTR + cc] = v;
    }
    __syncthreads();

    // planes: 0-2 guide, 3-6 src, 7-18 Ip(i*4+c), 19-24 II pairs
    for (int p = wave; p < 25; p += 4) {
        auto srcAt = [&](int k, int n) -> float {
            const int o = k * HSTR + n;
            if (p < 3)  return Gs[p][o];
            if (p < 7)  return Ss[p - 3][o];
            if (p < 19) { const int q = p - 7; return Gs[q >> 2][o] * Ss[q & 3][o]; }
            const int q = p - 19;                       // (0,0)(0,1)(0,2)(1,1)(1,2)(2,2)
            const int i = (q < 3) ? 0 : ((q < 5) ? 1 : 2);
            const int j = (q < 3) ? q : ((q < 5) ? q - 2 : 2);
            return Gs[i][o] * Gs[j][o];
        };
        v8f d = box2d_wmma(lane, Ts[wave], srcAt);
#pragma unroll
        for (int v = 0; v < 8; ++v)
            Ms[p][(v + half * 8) * TILE + l15] = d[v];
    }
    __syncthreads();

    // per-pixel covariance solve: 2 pixels per thread
    for (int pp = tid; pp < TILE * TILE; pp += 128) {
        const int py = pp >> 4, px = pp & 15;
        const int y  = ty0 + py, xg = tx0 + px;
        const float invN = 1.f / (nlen(y) * nlen(xg));

        float mI[3], mP[4];
#pragma unroll
        for (int i = 0; i < 3; ++i) mI[i] = Ms[i][pp] * invN;
#pragma unroll
        for (int c = 0; c < 4; ++c) mP[c] = Ms[3 + c][pp] * invN;

        float cov[3][4];
#pragma unroll
        for (int i = 0; i < 3; ++i)
#pragma unroll
            for (int c = 0; c < 4; ++c)
                cov[i][c] = Ms[7 + i * 4 + c][pp] * invN - mI[i] * mP[c];

        const float rr = Ms[19][pp] * invN - mI[0] * mI[0] + EPSV;
        const float rg = Ms[20][pp] * invN - mI[0] * mI[1];
        const float rb = Ms[21][pp] * invN - mI[0] * mI[2];
        const float gg = Ms[22][pp] * invN - mI[1] * mI[1] + EPSV;
        const float gb = Ms[23][pp] * invN - mI[1] * mI[2];
        const float bb = Ms[24][pp] * invN - mI[2] * mI[2] + EPSV;

        const float det = rr * gg * bb + 2.f * rg * gb * rb
                        - rb * gg * rb - rg * rg * bb - rr * gb * gb;
        const float id  = 1.f / det;
        const float i00 =  (gg * bb - gb * gb) * id;
        const float i01 = -(rg * bb - rb * gb) * id;
        const float i02 =  (rg * gb - rb * gg) * id;
        const float i11 =  (rr * bb - rb * rb) * id;
        const float i12 = -(rr * gb - rb * rg) * id;
        const float i22 =  (rr * gg - rg * rg) * id;
        const float inv[3][3] = {{i00, i01, i02}, {i01, i11, i12}, {i02, i12, i22}};

        const size_t base = ((size_t)bz * 16) * HW + (size_t)y * WD + xg;
        float A2[3][4];
#pragma unroll
        for (int j = 0; j < 3; ++j)
#pragma unroll
            for (int c = 0; c < 4; ++c) {
                A2[j][c] = cov[0][c] * inv[0][j] + cov[1][c] * inv[1][j]
                         + cov[2][c] * inv[2][j];
                ab[base + (size_t)(j * 4 + c) * HW] = A2[j][c];
            }
#pragma unroll
        for (int c = 0; c < 4; ++c) {
            const float bc = mP[c] - (A2[0][c] * mI[0] + A2[1][c] * mI[1]
                                    + A2[2][c] * mI[2]);
            ab[base + (size_t)(12 + c) * HW] = bc;
        }
    }
}

// ---------------------------------------------------------------------------
// K2: wave w owns output channel c=w. Box-filter a[i][c] (i=0..2) and b[c]
//     via WMMA, fuse q = (sum_i box(a_ic)*guide_i + box(b_c)) / N.
// ---------------------------------------------------------------------------
__global__ __launch_bounds__(128) void gf_final(
        const float* __restrict__ ab, const float* __restrict__ g,
        float* __restrict__ out) {
    __shared__ float Gt[3][TILE * TILE];   // guide at output pixels
    __shared__ float Hs[4][HALO * HSTR];   // per-wave halo buffer
    __shared__ float Ts[4][TILE * HSTR];   // per-wave WMMA scratch

    const int tid  = threadIdx.x;
    const int lane = tid & 31;
    const int wave = tid >> 5;             // == output channel c
    const int half = lane >> 4;
    const int l15  = lane & 15;
    const int tx0  = blockIdx.x * TILE;
    const int ty0  = blockIdx.y * TILE;
    const int bz   = blockIdx.z;

    const float* gp = g + (size_t)bz * 3 * HW;

    for (int idx = tid; idx < 3 * TILE * TILE; idx += 128) {
        const int p  = idx >> 8;
        const int pp = idx & 255;
        const int y  = ty0 + (pp >> 4);
        const int xg = tx0 + (pp & 15);
        Gt[p][pp] = gp[(size_t)p * HW + y * WD + xg];
    }
    __syncthreads();

    v8f qacc = {0.f, 0.f, 0.f, 0.f, 0.f, 0.f, 0.f, 0.f};
#pragma unroll
    for (int i = 0; i < 4; ++i) {
        const int plane = (i < 3) ? (i * 4 + wave) : (12 + wave);
        const float* src = ab + ((size_t)bz * 16 + plane) * HW;
        // wave-cooperative zero-padded halo load (in-order DS handles reuse)
        for (int idx = lane; idx < HALO * HALO; idx += 32) {
            const int r  = idx >> 5;
            const int cc = idx & 31;
            const int gy = ty0 - RAD + r;
            const int gx = tx0 - RAD + cc;
            float v = 0.f;
            if (gy >= 0 && gy < HD && gx >= 0 && gx < WD)
                v = src[(size_t)gy * WD + gx];
            Hs[wave][r * HSTR + cc] = v;
        }
        float* hb = Hs[wave];
        v8f d = box2d_wmma(lane, Ts[wave],
                           [&](int k, int n) -> float { return hb[k * HSTR + n]; });
        if (i < 3) {
#pragma unroll
            for (int v = 0; v < 8; ++v)
                qacc[v] += d[v] * Gt[i][(v + half * 8) * TILE + l15];
        } else {
#pragma unroll
            for (int v = 0; v < 8; ++v) qacc[v] += d[v];
        }
    }

#pragma unroll
    for (int v = 0; v < 8; ++v) {
        const int y  = ty0 + v + half * 8;
        const int xg = tx0 + l15;
        const float invN = 1.f / (nlen(y) * nlen(xg));
        out[((size_t)bz * 4 + wave) * HW + (size_t)y * WD + xg] = qacc[v] * invN;
    }
}

extern "C" void kernel_launch(void* const* d_in, const int* in_sizes, int n_in,
                              void* d_out, int out_size, void* d_ws, size_t ws_size,
                              hipStream_t stream) {
    const float* x = (const float*)d_in[0];   // (4,4,1024,1024)
    const float* g = (const float*)d_in[1];   // (4,3,1024,1024)
    float* ab  = (float*)d_ws;                // 16 planes * 4 batches = 256 MB
    float* out = (float*)d_out;

    dim3 grid(WD / TILE, HD / TILE, 4);
    dim3 block(128);
    gf_stats_solve<<<grid, block, 0, stream>>>(x, g, ab);
    gf_final<<<grid, block, 0, stream>>>(ab, g, out);
}